// TopKRouter_64673617543271
// MI455X (gfx1250) — compile-verified
//
#include <hip/hip_runtime.h>
#include <math.h>

typedef __attribute__((ext_vector_type(2))) float v2f;
typedef __attribute__((ext_vector_type(8))) float v8f;

#define T_TOKENS 8192
#define DM 4096
#define NE 64
#define KC 128            // k-chunk staged in LDS
#define WROW (KC + 4)     // +4 floats pad -> bank advance 4/row, conflict-free b64 reads
// workspace layout (floats): [0, T*64) logits ; then acc: imp[64], load[64], zsum, logitsum
#define ACC_OFF ((size_t)T_TOKENS * NE)

// ---------------- init accumulators ----------------
__global__ void router_init_acc(float* __restrict__ acc) {
  int i = threadIdx.x;
  if (i < 130) acc[i] = 0.0f;
}

// ---------------- GEMM: logits = x @ W^T via V_WMMA_F32_16X16X4_F32 ----------------
// Block = 8 waves; W k-chunk (64x128) staged once per block into LDS, B read via ds_load_b64.
// One wave -> 16 tokens x 64 experts (4 N-tiles of 16), K marched in steps of 4.
__global__ __launch_bounds__(256) void router_gemm_wmma(
    const float* __restrict__ x, const float* __restrict__ W,
    float* __restrict__ logits) {
  __shared__ float sW[64 * WROW];   // 33.8 KB of the WGP's 320 KB LDS

  const int tid  = (int)threadIdx.x;
  const int wave = (int)blockIdx.x * 8 + (tid >> 5);
  const int lane = tid & 31;
  const int half = lane >> 4;       // 0: K={0,1}, 1: K={2,3}
  const int r    = lane & 15;       // M row (A) / N col (B,C,D)
  const int tokenBase = wave * 16;

  // A: lane holds x[tokenBase+r][k0 + 2*half + {0,1}]
  const float* xrow = x + (size_t)(tokenBase + r) * DM + 2 * half;
  // B: lane reads sW[16n + r][k + 2*half + {0,1}]  (B = W^T tile)
  const float* bbase = &sW[(size_t)r * WROW + 2 * half];

  v8f acc0 = {}, acc1 = {}, acc2 = {}, acc3 = {};

  for (int kc = 0; kc < DM; kc += KC) {
    __syncthreads();
    // ---- stage W[:, kc:kc+KC] into LDS: 64*128 floats = 2048 float4, 8 per thread ----
    #pragma unroll
    for (int t = 0; t < 8; ++t) {
      int idx4 = tid + t * 256;         // 0..2047
      int row  = idx4 >> 5;             // 32 float4 per row
      int c4   = idx4 & 31;
      float4 v = *(const float4*)(W + (size_t)row * DM + kc + c4 * 4);
      *(float4*)(&sW[row * WROW + c4 * 4]) = v;
    }
    // prefetch next x chunk for this lane's token row while we compute this one
    if (kc + KC < DM) __builtin_prefetch(xrow + kc + KC, 0, 0);
    __syncthreads();

    #pragma unroll 4
    for (int k = 0; k < KC; k += 4) {
      v2f a  = *(const v2f*)(xrow + kc + k);           // global (streamed, HBM-bound)
      v2f b0 = *(const v2f*)(bbase + k);               // LDS, conflict-free
      v2f b1 = *(const v2f*)(bbase + 16 * WROW + k);
      v2f b2 = *(const v2f*)(bbase + 32 * WROW + k);
      v2f b3 = *(const v2f*)(bbase + 48 * WROW + k);
      acc0 = __builtin_amdgcn_wmma_f32_16x16x4_f32(false, a, false, b0, (short)0, acc0, false, false);
      acc1 = __builtin_amdgcn_wmma_f32_16x16x4_f32(false, a, false, b1, (short)0, acc1, false, false);
      acc2 = __builtin_amdgcn_wmma_f32_16x16x4_f32(false, a, false, b2, (short)0, acc2, false, false);
      acc3 = __builtin_amdgcn_wmma_f32_16x16x4_f32(false, a, false, b3, (short)0, acc3, false, false);
    }
  }

  // C/D layout: VGPR j holds row M=j (lanes 0-15) and M=j+8 (lanes 16-31), N=lane%16
  #pragma unroll
  for (int j = 0; j < 8; ++j) {
    int token = tokenBase + j + 8 * half;
    float* dst = logits + (size_t)token * NE + r;
    dst[0]  = acc0[j];
    dst[16] = acc1[j];
    dst[32] = acc2[j];
    dst[48] = acc3[j];
  }
}

// ---------------- softmax + top2 + partial losses (1 thread per token) ----------------
__global__ __launch_bounds__(256) void router_softmax_topk(
    const float* __restrict__ logits, float* __restrict__ out,
    float* __restrict__ acc) {
  __shared__ float s_imp[NE];
  __shared__ float s_load[NE];
  if (threadIdx.x < NE) { s_imp[threadIdx.x] = 0.0f; s_load[threadIdx.x] = 0.0f; }
  __syncthreads();

  const int t = (int)(blockIdx.x * blockDim.x + threadIdx.x);
  const float* row = logits + (size_t)t * NE;

  float m = -INFINITY;
  #pragma unroll 8
  for (int e = 0; e < NE; ++e) m = fmaxf(m, row[e]);

  float sum = 0.0f, lsum = 0.0f;
  float b1 = -INFINITY, b2 = -INFINITY;
  int i1 = 0, i2 = 0;
  #pragma unroll 8
  for (int e = 0; e < NE; ++e) {
    float v = row[e];
    lsum += v;
    sum  += __expf(v - m);
    if (v > b1)      { b2 = b1; i2 = i1; b1 = v; i1 = e; }
    else if (v > b2) { b2 = v; i2 = e; }
  }
  const float inv = 1.0f / sum;
  const float p1 = __expf(b1 - m) * inv;
  const float p2 = __expf(b2 - m) * inv;
  const float denom = fmaxf(p1 + p2, 1e-9f);

  out[(size_t)t * 2]     = (float)i1;          // topk_experts
  out[(size_t)t * 2 + 1] = (float)i2;
  float* scores = out + (size_t)2 * T_TOKENS;
  scores[(size_t)t * 2]     = p1 / denom;      // topk_scores (normalized)
  scores[(size_t)t * 2 + 1] = p2 / denom;

  const float lse = m + __logf(sum);
  atomicAdd(&acc[128], lse * lse);             // z-loss partial
  atomicAdd(&acc[129], lsum);                  // logits-sum partial

  #pragma unroll 8
  for (int e = 0; e < NE; ++e) atomicAdd(&s_imp[e], __expf(row[e] - m) * inv);
  atomicAdd(&s_load[i1], 1.0f);
  __syncthreads();
  if (threadIdx.x < NE) {
    atomicAdd(&acc[threadIdx.x],      s_imp[threadIdx.x]);
    atomicAdd(&acc[NE + threadIdx.x], s_load[threadIdx.x]);
  }
}

// ---------------- finalize scalars ----------------
__global__ void router_finalize(const float* __restrict__ acc, float* __restrict__ out) {
  if (threadIdx.x == 0 && blockIdx.x == 0) {
    float impTot = 0.0f, loadTot = 0.0f;
    for (int e = 0; e < NE; ++e) { impTot += acc[e]; loadTot += acc[NE + e]; }
    impTot  = fmaxf(impTot, 1e-9f);
    loadTot = fmaxf(loadTot, 1e-9f);
    float lb = 0.0f;
    for (int e = 0; e < NE; ++e) lb += (acc[e] / impTot) * (acc[NE + e] / loadTot);
    lb *= (float)(NE * NE) * 0.01f;
    out[(size_t)4 * T_TOKENS]     = acc[128] / (float)T_TOKENS * 0.001f;  // router_z_loss
    out[(size_t)4 * T_TOKENS + 1] = lb;                                   // load_balance_loss
    out[(size_t)4 * T_TOKENS + 2] = acc[129] / ((float)T_TOKENS * NE);    // logits_mean
  }
}

extern "C" void kernel_launch(void* const* d_in, const int* in_sizes, int n_in,
                              void* d_out, int out_size, void* d_ws, size_t ws_size,
                              hipStream_t stream) {
  const float* x = (const float*)d_in[0];   // (8192, 4096) f32
  const float* W = (const float*)d_in[1];   // (64, 4096)   f32
  float* out    = (float*)d_out;            // 16384 experts + 16384 scores + 3 scalars
  float* logits = (float*)d_ws;             // 8192*64 f32
  float* acc    = logits + ACC_OFF;         // imp[64], load[64], zsum, lsum

  router_init_acc<<<1, 256, 0, stream>>>(acc);
  // 512 waves, each owns 16 tokens x 64 experts; 8 waves/block share the LDS W-chunk
  router_gemm_wmma<<<T_TOKENS / 16 / 8, 256, 0, stream>>>(x, W, logits);
  router_softmax_topk<<<T_TOKENS / 256, 256, 0, stream>>>(logits, out, acc);
  router_finalize<<<1, 64, 0, stream>>>(acc, out);
}